// CNN_24438363914287
// MI455X (gfx1250) — compile-verified
//
#include <hip/hip_runtime.h>
#include <hip/hip_bf16.h>

// Problem constants (from reference)
#define VE   300      // E: embedding dim
#define TT   50       // T: tags
#define FF   100      // F: conv filters
#define KK   3        // conv kernel
#define BB   32       // batch
#define SS   512      // seq len
#define EK   900      // E*K (im2col depth)
#define FP   112      // F padded to 7*16
#define TP   64       // T padded to 4*16

#define MR   32       // M rows per workgroup (two 16-row WMMA tiles)
#define NROW 34       // gathered emb rows per tile: positions s0-1 .. s0+32
#define LDR  308      // emb-row LDS stride: ==4 mod 8 -> conflict-free, b64-aligned
#define LDF  116      // feat LDS stride:    ==4 mod 8 -> conflict-free, b64-aligned

typedef float v2f __attribute__((ext_vector_type(2)));
typedef float v8f __attribute__((ext_vector_type(8)));

__device__ __forceinline__ v8f wmma_f32_4(v2f a, v2f b, v8f c) {
    // V_WMMA_F32_16X16X4_F32 : D = A(16x4) * B(4x16) + C(16x16), all f32
    return __builtin_amdgcn_wmma_f32_16x16x4_f32(
        /*neg_a=*/false, a, /*neg_b=*/false, b,
        /*c_mod=*/(short)0, c, /*reuse_a=*/false, /*reuse_b=*/false);
}

// ---------------------------------------------------------------------------
// Kernel 0: repack weights into pair-interleaved, zero-padded B layouts so a
// lane's two K values are one contiguous 8-byte load.
//   W1p[(kk>>1)][f][kk&1] = conv_w[f, e, kc]   with kk = kc*E + e
//   W2p[(f>>1)][t][f&1]   = fc_w[t, f]
// ---------------------------------------------------------------------------
__global__ void pack_weights(const float* __restrict__ conv_w,
                             const float* __restrict__ fc_w,
                             float* __restrict__ W1p,
                             float* __restrict__ W2p) {
    int i = blockIdx.x * blockDim.x + threadIdx.x;
    if (i < EK * FP) {
        int kk = i / FP;
        int f  = i - kk * FP;
        float v = 0.f;
        if (f < FF) {
            int kc = kk / VE;
            int e  = kk - kc * VE;
            v = conv_w[f * EK + e * KK + kc];
        }
        W1p[(kk >> 1) * (FP * 2) + f * 2 + (kk & 1)] = v;
    }
    if (i < FP * TP) {
        int ff = i / TP;
        int t  = i - ff * TP;
        float v = (t < TT && ff < FF) ? fc_w[t * FF + ff] : 0.f;
        W2p[(ff >> 1) * (TP * 2) + t * 2 + (ff & 1)] = v;
    }
}

// ---------------------------------------------------------------------------
// Kernel 1: fused gather + conv-GEMM + ReLU + FC-GEMM -> emissions.
// One wave per 32-row M tile; the conv's im2col is virtual: A[m, kc*300+e]
// reads gathered row (m+kc), so each emb row is stored once.
// Both 16-row sub-tiles share every B fragment (2 WMMAs per b64 load).
// ---------------------------------------------------------------------------
__global__ __launch_bounds__(32)
void emissions_kernel(const int*   __restrict__ x,
                      const float* __restrict__ emb,
                      const float* __restrict__ W1p,
                      const float* __restrict__ conv_b,
                      const float* __restrict__ W2p,
                      const float* __restrict__ fc_b,
                      float* __restrict__ em) {
    __shared__ float rows[NROW * LDR];   // gathered emb rows (pos s0-1 .. s0+32)
    __shared__ float feat[MR * LDF];     // conv+ReLU output tile
    __shared__ int   srcRow[NROW];       // emb row index or -1 = zero pad

    const int lane = threadIdx.x;
    const int tile = blockIdx.x;         // 0 .. B*S/32 - 1
    const int p0   = tile * MR;          // first global row (32 | 512 -> single batch)
    const int b    = p0 >> 9;
    const int s0   = p0 & 511;

    // --- gather source row indices (boundary -> -1) ---
    for (int r = lane; r < NROW; r += 32) {
        int s = s0 + r - 1;
        srcRow[r] = (s >= 0 && s < SS) ? x[b * SS + s] : -1;
    }
    __syncthreads();

    // --- stage 34 distinct emb rows in LDS (no im2col duplication) ---
    for (int i = lane; i < NROW * VE; i += 32) {
        int r   = i / VE;
        int e   = i - r * VE;
        int row = srcRow[r];
        rows[r * LDR + e] = (row >= 0) ? emb[row * VE + e] : 0.f;
    }
    __syncthreads();

    const int m     = lane & 15;           // A-row index within sub-tile
    const int n     = lane & 15;           // B/C column index
    const int koff  = (lane >> 4) << 1;    // 0 (lanes 0-15) or 2 (lanes 16-31)
    const int khalf = lane >> 4;           // pair offset: 0 or 1
    const int mhi   = (lane >> 4) << 3;    // C rows: v + 0 or v + 8

    // --- GEMM1: [32 x 900] x [900 x 112] via f32 WMMA, fused bias + ReLU ---
    for (int nt = 0; nt < 7; ++nt) {
        const int f = nt * 16 + n;
        const float* bcol = W1p + f * 2;   // + pair*(FP*2)
        v8f acc0 = {}, acc1 = {};
        for (int kc = 0; kc < KK; ++kc) {  // virtual im2col: segment kc
            const float* a0row = &rows[(m + kc) * LDR];
            const float* a1row = &rows[(m + 16 + kc) * LDR];
            const float* bseg  = &bcol[(((kc * VE) >> 1) + khalf) * (FP * 2)];
#pragma unroll 2
            for (int k0 = 0; k0 < VE; k0 += 4) {
                v2f a0 = *(const v2f*)&a0row[k0 + koff];     // ds_load_b64
                v2f a1 = *(const v2f*)&a1row[k0 + koff];     // ds_load_b64
                v2f bb = *(const v2f*)&bseg[(k0 >> 1) * (FP * 2)]; // global_load_b64
                acc0 = wmma_f32_4(a0, bb, acc0);
                acc1 = wmma_f32_4(a1, bb, acc1);
            }
        }
        const float bias = (f < FF) ? conv_b[f] : 0.f;
#pragma unroll
        for (int v = 0; v < 8; ++v) {
            int mr = v + mhi;
            feat[mr * LDF + nt * 16 + n]        = fmaxf(acc0[v] + bias, 0.f);
            feat[(mr + 16) * LDF + nt * 16 + n] = fmaxf(acc1[v] + bias, 0.f);
        }
    }
    __syncthreads();

    // --- GEMM2: [32 x 112] x [112 x 64] via f32 WMMA, fused fc_b ---
    for (int nt = 0; nt < 4; ++nt) {
        const int t = nt * 16 + n;
        const float* bcol = W2p + t * 2;
        v8f acc0 = {}, acc1 = {};
#pragma unroll 2
        for (int k0 = 0; k0 < FP; k0 += 4) {
            v2f a0 = *(const v2f*)&feat[m * LDF + k0 + koff];
            v2f a1 = *(const v2f*)&feat[(m + 16) * LDF + k0 + koff];
            v2f bb = *(const v2f*)&bcol[((k0 >> 1) + khalf) * (TP * 2)];
            acc0 = wmma_f32_4(a0, bb, acc0);
            acc1 = wmma_f32_4(a1, bb, acc1);
        }
        if (t < TT) {
            const float bias = fc_b[t];
#pragma unroll
            for (int v = 0; v < 8; ++v) {
                int mr = v + mhi;
                em[(p0 + mr) * TT + t]      = acc0[v] + bias;
                em[(p0 + mr + 16) * TT + t] = acc1[v] + bias;
            }
        }
    }
}

// ---------------------------------------------------------------------------
// Kernel 2: CRF forward algorithm + gold-path score, one block per batch.
// Per-thread register cache of the trans column, tree reductions, and a
// double-buffered alpha (one barrier per time step).
// ---------------------------------------------------------------------------
__global__ __launch_bounds__(64)
void crf_kernel(const float* __restrict__ em,
                const int*   __restrict__ tags,
                const float* __restrict__ start_trans,
                const float* __restrict__ end_trans,
                const float* __restrict__ trans,
                float* __restrict__ batch_nll) {
    __shared__ float s_trans[TT * TT];
    __shared__ float alpha[2][TT];
    __shared__ float red[64];

    const int b = blockIdx.x;
    const int j = threadIdx.x;

    for (int i = j; i < TT * TT; i += 64) s_trans[i] = trans[i];
    const float* emB = em + (size_t)b * SS * TT;

    float tcol[TT];                       // trans[:, j] cached in registers
    if (j < TT) {
#pragma unroll
        for (int i = 0; i < TT; ++i) tcol[i] = trans[i * TT + j];
        alpha[0][j] = start_trans[j] + emB[j];
    }
    __syncthreads();

    // sequential forward scan over time (double-buffered: 1 barrier/step)
    for (int s = 1; s < SS; ++s) {
        const float* ap = alpha[(s - 1) & 1];
        if (j < TT) {
            float av[TT];
#pragma unroll
            for (int i = 0; i < TT; ++i) av[i] = ap[i] + tcol[i];

            float mv[64];                 // max tree (pad with -huge)
#pragma unroll
            for (int i = 0; i < 64; ++i) mv[i] = (i < TT) ? av[i] : -3.4e38f;
#pragma unroll
            for (int st = 32; st >= 1; st >>= 1)
#pragma unroll
                for (int i = 0; i < st; ++i) mv[i] = fmaxf(mv[i], mv[i + st]);
            const float mx = mv[0];

#pragma unroll
            for (int i = 0; i < 64; ++i) mv[i] = (i < TT) ? __expf(av[i] - mx) : 0.f;
#pragma unroll
            for (int st = 32; st >= 1; st >>= 1)
#pragma unroll
                for (int i = 0; i < st; ++i) mv[i] += mv[i + st];

            alpha[s & 1][j] = mx + __logf(mv[0]) + emB[s * TT + j];
        }
        __syncthreads();
    }
    const float* af = alpha[(SS - 1) & 1];

    // gold-path partial sums (emissions + transitions), strided over s
    const int* tg = tags + b * SS;
    float part = 0.f;
    for (int s = j; s < SS; s += 64) {
        int t1 = tg[s];
        part += emB[s * TT + t1];
        if (s > 0) part += s_trans[tg[s - 1] * TT + t1];
    }
    red[j] = part;
    __syncthreads();

    if (j == 0) {
        // log Z = logsumexp_j(alpha[j] + end_trans[j])
        float mx = -3.4e38f;
        for (int i = 0; i < TT; ++i) mx = fmaxf(mx, af[i] + end_trans[i]);
        float sum = 0.f;
        for (int i = 0; i < TT; ++i) sum += __expf(af[i] + end_trans[i] - mx);
        float logz = mx + __logf(sum);

        float num = start_trans[tg[0]] + end_trans[tg[SS - 1]];
        for (int i = 0; i < 64; ++i) num += red[i];
        batch_nll[b] = logz - num;       // per-batch NLL
    }
}

// ---------------------------------------------------------------------------
// Kernel 3: sum per-batch NLLs -> scalar output.
// ---------------------------------------------------------------------------
__global__ void final_reduce(const float* __restrict__ batch_nll,
                             float* __restrict__ out) {
    if (threadIdx.x == 0 && blockIdx.x == 0) {
        float s = 0.f;
        for (int i = 0; i < BB; ++i) s += batch_nll[i];
        out[0] = s;
    }
}

// ---------------------------------------------------------------------------
extern "C" void kernel_launch(void* const* d_in, const int* in_sizes, int n_in,
                              void* d_out, int out_size, void* d_ws, size_t ws_size,
                              hipStream_t stream) {
    const int*   x        = (const int*)  d_in[0];
    const int*   tags     = (const int*)  d_in[1];
    const float* emb      = (const float*)d_in[2];
    const float* conv_w   = (const float*)d_in[3];
    const float* conv_b   = (const float*)d_in[4];
    const float* fc_w     = (const float*)d_in[5];
    const float* fc_b     = (const float*)d_in[6];
    const float* start_tr = (const float*)d_in[7];
    const float* end_tr   = (const float*)d_in[8];
    const float* trans    = (const float*)d_in[9];

    // workspace layout (floats)
    float* em   = (float*)d_ws;                       // B*S*T      = 819,200
    float* W1p  = em  + (size_t)BB * SS * TT;         // EK*FP      = 100,800
    float* W2p  = W1p + (size_t)EK * FP;              // FP*TP      =   7,168
    float* bnll = W2p + (size_t)FP * TP;              // B          =      32

    pack_weights   <<<(EK * FP + 255) / 256, 256, 0, stream>>>(conv_w, fc_w, W1p, W2p);
    emissions_kernel<<<(BB * SS) / MR,        32, 0, stream>>>(x, emb, W1p, conv_b, W2p, fc_b, em);
    crf_kernel     <<<BB,                     64, 0, stream>>>(em, tags, start_tr, end_tr, trans, bnll);
    final_reduce   <<<1,                      32, 0, stream>>>(bnll, (float*)d_out);
}